// ClassicGNN_31705448579347
// MI455X (gfx1250) — compile-verified
//
#include <hip/hip_runtime.h>

#define N_NODES 100000
#define N_EDGES 1600000
#define DIM 128
#define NEG_SLOPE 0.01f
#define BN_EPS 1e-5f
#define GPAD 132  // padded LDS row stride (floats)

typedef __attribute__((ext_vector_type(2))) float v2f;
typedef __attribute__((ext_vector_type(8))) float v8f;

__device__ __forceinline__ float lrelu(float x) { return x > 0.f ? x : NEG_SLOPE * x; }

// x[n][:] = embed[node_deg[n]][:]  (one float4 per thread, 32 threads per row)
__global__ void embed_kernel(const int* __restrict__ deg, const float* __restrict__ embed,
                             float* __restrict__ x) {
  int t = blockIdx.x * blockDim.x + threadIdx.x;
  int n = t >> 5;
  int c4 = (t & 31) << 2;
  if (n >= N_NODES) return;
  float4 v = *(const float4*)(embed + (size_t)deg[n] * DIM + c4);
  *(float4*)(x + (size_t)n * DIM + c4) = v;
}

// agg = (1 + eps[layer]) * x
__global__ void init_agg_kernel(const float* __restrict__ x, float* __restrict__ agg,
                                const float* __restrict__ eps, int layer) {
  float s = 1.0f + eps[layer];
  size_t t = (size_t)blockIdx.x * blockDim.x + threadIdx.x;
  if (t >= (size_t)N_NODES * DIM / 4) return;
  float4 v = *(const float4*)(x + t * 4);
  v.x *= s; v.y *= s; v.z *= s; v.w *= s;
  *(float4*)(agg + t * 4) = v;
}

// one wave per edge: agg[dst] += x[src]  (float4 per lane, 4 f32 atomics)
__global__ void scatter_kernel(const int* __restrict__ ei, const float* __restrict__ x,
                               float* __restrict__ agg) {
  int wave = (blockIdx.x * blockDim.x + threadIdx.x) >> 5;
  int lane = threadIdx.x & 31;
  if (wave >= N_EDGES) return;
  int s = ei[wave];
  int d = ei[N_EDGES + wave];
  float4 v = *(const float4*)(x + (size_t)s * DIM + lane * 4);
  float* o = agg + (size_t)d * DIM + lane * 4;
  atomicAdd(o + 0, v.x);
  atomicAdd(o + 1, v.y);
  atomicAdd(o + 2, v.z);
  atomicAdd(o + 3, v.w);
}

// Y = [lrelu](X @ W + bias); optionally accumulate per-column sum / sum^2.
// Block: 256 threads = 8 waves; 32 rows per block (N_NODES % 32 == 0);
// wave w computes the 16-column tile [w*16, w*16+16) for both 16-row tiles.
__global__ void gin_gemm_kernel(const float* __restrict__ X, const float* __restrict__ W,
                                const float* __restrict__ bias, float* __restrict__ Y,
                                int apply_lrelu, float* __restrict__ colsum,
                                float* __restrict__ colsumsq, int do_stats) {
  __shared__ float As[32 * GPAD];
  const int tid = threadIdx.x;
  const int row0 = blockIdx.x * 32;

  // stage 32x128 A tile into LDS: each thread moves 16 contiguous floats
  {
    int r = tid >> 3;
    int c0 = (tid & 7) * 16;
    const float4* src = (const float4*)(X + (size_t)(row0 + r) * DIM + c0);
    float4 a0 = src[0], a1 = src[1], a2 = src[2], a3 = src[3];
    float* dst = As + r * GPAD + c0;
    *(float4*)(dst + 0) = a0;
    *(float4*)(dst + 4) = a1;
    *(float4*)(dst + 8) = a2;
    *(float4*)(dst + 12) = a3;
  }
  __syncthreads();

  const int wave = tid >> 5;
  const int lane = tid & 31;
  const int m = lane & 15;               // row within tile
  const int col = wave * 16 + m;         // output column
  const int khalf = (lane >> 4) * 2;     // K offset: 0 (lanes 0-15) or 2 (lanes 16-31)

  v8f acc0 = {0.f, 0.f, 0.f, 0.f, 0.f, 0.f, 0.f, 0.f};
  v8f acc1 = {0.f, 0.f, 0.f, 0.f, 0.f, 0.f, 0.f, 0.f};

#pragma unroll 8
  for (int k4 = 0; k4 < DIM; k4 += 4) {
    int k = k4 + khalf;
    v2f b;
    b.x = W[(size_t)k * DIM + col];
    b.y = W[(size_t)(k + 1) * DIM + col];
    v2f a0;
    a0.x = As[m * GPAD + k];
    a0.y = As[m * GPAD + k + 1];
    v2f a1;
    a1.x = As[(m + 16) * GPAD + k];
    a1.y = As[(m + 16) * GPAD + k + 1];
    acc0 = __builtin_amdgcn_wmma_f32_16x16x4_f32(false, a0, false, b, (short)0, acc0,
                                                 false, false);
    acc1 = __builtin_amdgcn_wmma_f32_16x16x4_f32(false, a1, false, b, (short)0, acc1,
                                                 false, false);
  }

  const float bcol = bias[col];
  float s = 0.f, s2 = 0.f;
  const int rbase = row0 + (lane >> 4) * 8;  // C/D layout: VGPR r -> M=r (lanes<16), M=r+8
#pragma unroll
  for (int r = 0; r < 8; ++r) {
    float y0 = acc0[r] + bcol;
    float y1 = acc1[r] + bcol;
    if (apply_lrelu) { y0 = lrelu(y0); y1 = lrelu(y1); }
    Y[(size_t)(rbase + r) * DIM + col] = y0;
    Y[(size_t)(rbase + 16 + r) * DIM + col] = y1;
    s += y0 + y1;
    s2 += y0 * y0 + y1 * y1;
  }

  if (do_stats) {
    s += __shfl_xor(s, 16, 32);
    s2 += __shfl_xor(s2, 16, 32);
    if (lane < 16) {  // each wave owns distinct columns -> no intra-block collisions
      atomicAdd(&colsum[col], s);
      atomicAdd(&colsumsq[col], s2);
    }
  }
}

__global__ void zero_stats_kernel(float* __restrict__ stats) { stats[threadIdx.x] = 0.f; }

__global__ void bn_finalize_kernel(const float* __restrict__ colsum,
                                   const float* __restrict__ colsumsq,
                                   float* __restrict__ mean, float* __restrict__ rsig) {
  int c = threadIdx.x;  // 128 threads
  float inv_n = 1.0f / (float)N_NODES;
  float mu = colsum[c] * inv_n;
  float var = colsumsq[c] * inv_n - mu * mu;
  mean[c] = mu;
  rsig[c] = rsqrtf(var + BN_EPS);
}

// h = lrelu((h - mean) * rsig * gamma + beta), in place, float4-vectorized
__global__ void bn_apply_kernel(float* __restrict__ h, const float* __restrict__ mean,
                                const float* __restrict__ rsig,
                                const float* __restrict__ gamma,
                                const float* __restrict__ beta) {
  size_t t = (size_t)blockIdx.x * blockDim.x + threadIdx.x;
  if (t >= (size_t)N_NODES * DIM / 4) return;
  int c4 = (int)(t & 31) * 4;
  float4 v = *(float4*)(h + t * 4);
  float4 mu = *(const float4*)(mean + c4);
  float4 rs = *(const float4*)(rsig + c4);
  float4 g = *(const float4*)(gamma + c4);
  float4 b = *(const float4*)(beta + c4);
  v.x = lrelu((v.x - mu.x) * rs.x * g.x + b.x);
  v.y = lrelu((v.y - mu.y) * rs.y * g.y + b.y);
  v.z = lrelu((v.z - mu.z) * rs.z * g.z + b.z);
  v.w = lrelu((v.w - mu.w) * rs.w * g.w + b.w);
  *(float4*)(h + t * 4) = v;
}

extern "C" void kernel_launch(void* const* d_in, const int* in_sizes, int n_in,
                              void* d_out, int out_size, void* d_ws, size_t ws_size,
                              hipStream_t stream) {
  const int* node_deg = (const int*)d_in[0];
  const int* edge_index = (const int*)d_in[1];
  const float* embed = (const float*)d_in[2];
  const float* W1 = (const float*)d_in[3];
  const float* b1 = (const float*)d_in[4];
  const float* W2 = (const float*)d_in[5];
  const float* b2 = (const float*)d_in[6];
  const float* eps = (const float*)d_in[7];
  const float* gamma = (const float*)d_in[8];
  const float* beta = (const float*)d_in[9];
  float* out = (float*)d_out;

  const size_t ND = (size_t)N_NODES * DIM;
  float* bufX = (float*)d_ws;       // current node features
  float* bufAgg = bufX + ND;        // (1+eps)x + neighbor sum
  float* bufH = bufAgg + ND;        // hidden after GEMM1
  float* stats = bufH + ND;         // colsum[128] colsumsq[128] mean[128] rsig[128]
  float* colsum = stats;
  float* colsumsq = stats + 128;
  float* mean = stats + 256;
  float* rsig = stats + 384;

  const int elem4_blocks = (int)((ND / 4 + 255) / 256);

  embed_kernel<<<(N_NODES * 32 + 255) / 256, 256, 0, stream>>>(node_deg, embed, bufX);

  for (int i = 0; i < 4; ++i) {
    init_agg_kernel<<<elem4_blocks, 256, 0, stream>>>(bufX, bufAgg, eps, i);
    scatter_kernel<<<(N_EDGES * 32 + 255) / 256, 256, 0, stream>>>(edge_index, bufX, bufAgg);
    gin_gemm_kernel<<<N_NODES / 32, 256, 0, stream>>>(
        bufAgg, W1 + (size_t)i * DIM * DIM, b1 + i * DIM, bufH,
        /*lrelu=*/1, nullptr, nullptr, /*stats=*/0);
    if (i < 3) {
      zero_stats_kernel<<<1, 256, 0, stream>>>(stats);
      gin_gemm_kernel<<<N_NODES / 32, 256, 0, stream>>>(
          bufH, W2 + (size_t)i * DIM * DIM, b2 + i * DIM, bufX,
          /*lrelu=*/0, colsum, colsumsq, /*stats=*/1);
      bn_finalize_kernel<<<1, 128, 0, stream>>>(colsum, colsumsq, mean, rsig);
      bn_apply_kernel<<<elem4_blocks, 256, 0, stream>>>(bufX, mean, rsig,
                                                        gamma + i * DIM, beta + i * DIM);
    } else {
      gin_gemm_kernel<<<N_NODES / 32, 256, 0, stream>>>(
          bufH, W2 + (size_t)i * DIM * DIM, b2 + i * DIM, out,
          /*lrelu=*/0, nullptr, nullptr, /*stats=*/0);
    }
  }
}